// NodeClassifier_85796266705312
// MI455X (gfx1250) — compile-verified
//
#include <hip/hip_runtime.h>
#include <hip/hip_bf16.h>

#define NN     50000
#define NRELC  20
#define RR     41          // 2*NREL + 1
#define NHIDC  16
#define NCLS   50
#define KTOT   656         // RR * NHIDC
#define NCPAD  64          // NCLS padded to 4 WMMA column tiles

typedef float v2f __attribute__((ext_vector_type(2)));
typedef float v8f __attribute__((ext_vector_type(8)));

// ---------------- init kernels ----------------

__global__ void zero4_kernel(float4* __restrict__ p, long long n4) {
    long long i = (long long)blockIdx.x * blockDim.x + threadIdx.x;
    long long stride = (long long)gridDim.x * blockDim.x;
    for (; i < n4; i += stride) p[i] = make_float4(0.f, 0.f, 0.f, 0.f);
}

// h[n,k] = b1[k] + W1[2*NREL, n, k]   (self-loop edge has val == 1, folded in here)
__global__ void hinit_kernel(float* __restrict__ h, const float* __restrict__ b1,
                             const float* __restrict__ W1) {
    int t = blockIdx.x * blockDim.x + threadIdx.x;
    if (t >= NN * NHIDC) return;
    int k = t & (NHIDC - 1);
    h[t] = b1[k] + W1[(size_t)(2 * NRELC) * NN * NHIDC + t];
}

// W2p[k, c] = (c < 50) ? W2[k, c] : 0   with k = r*16+h flattened, c padded to 64
__global__ void w2pad_kernel(float* __restrict__ w2p, const float* __restrict__ W2) {
    int t = blockIdx.x * blockDim.x + threadIdx.x;
    if (t >= KTOT * NCPAD) return;
    int c = t & (NCPAD - 1);
    int k = t >> 6;
    w2p[t] = (c < NCLS) ? W2[(size_t)k * NCLS + c] : 0.0f;
}

// ---------------- degree (per (relation, source) row counts) ----------------

__global__ void deg_kernel(const int* __restrict__ src, const int* __restrict__ rel,
                           const int* __restrict__ dst, float* __restrict__ deg, int E) {
    int i = blockIdx.x * blockDim.x + threadIdx.x;
    if (i >= E) return;
    int r = rel[i];
    atomicAdd(deg + (size_t)r * NN + src[i], 1.0f);
    atomicAdd(deg + (size_t)(r + NRELC) * NN + dst[i], 1.0f);
}

// ---------------- layer 1: h[s] += val * W1[r, o, :]  (non-self edges) ----------------

__global__ void l1_kernel(const int* __restrict__ src, const int* __restrict__ rel,
                          const int* __restrict__ dst, const float* __restrict__ W1,
                          const float* __restrict__ deg, float* __restrict__ h, int E) {
    long long j = (long long)blockIdx.x * blockDim.x + threadIdx.x;
    if (j >= 2LL * E) return;
    int s, o, r;
    if (j < E) { int i = (int)j;       s = src[i]; o = dst[i]; r = rel[i]; }
    else       { int i = (int)(j - E); s = dst[i]; o = src[i]; r = rel[i] + NRELC; }
    size_t row = (size_t)r * NN;
    float val = 1.0f / deg[row + s];
    const float4* w = (const float4*)(W1 + (row + o) * NHIDC);
    float* hs = h + (size_t)s * NHIDC;
#pragma unroll
    for (int q = 0; q < 4; ++q) {
        float4 wv = w[q];
        atomicAdd(hs + q * 4 + 0, val * wv.x);
        atomicAdd(hs + q * 4 + 1, val * wv.y);
        atomicAdd(hs + q * 4 + 2, val * wv.z);
        atomicAdd(hs + q * 4 + 3, val * wv.w);
    }
}

// ---------------- layer 2 scatter: af2[s, r*16 + k] += val * relu(h[o, k]) ----------------

__global__ void l2_kernel(const int* __restrict__ src, const int* __restrict__ rel,
                          const int* __restrict__ dst, const float* __restrict__ deg,
                          const float* __restrict__ h, float* __restrict__ af2, int E) {
    long long j = (long long)blockIdx.x * blockDim.x + threadIdx.x;
    long long tot = 2LL * E + NN;
    if (j >= tot) return;
    int s, o, r;
    float val;
    if (j < 2LL * E) {
        if (j < E) { int i = (int)j;       s = src[i]; o = dst[i]; r = rel[i]; }
        else       { int i = (int)(j - E); s = dst[i]; o = src[i]; r = rel[i] + NRELC; }
        val = 1.0f / deg[(size_t)r * NN + s];
    } else {  // self loop: deg row has exactly one edge -> val = 1
        int n = (int)(j - 2LL * E);
        s = n; o = n; r = 2 * NRELC; val = 1.0f;
    }
    const float4* hp = (const float4*)(h + (size_t)o * NHIDC);
    float* ap = af2 + (size_t)s * KTOT + r * NHIDC;
#pragma unroll
    for (int q = 0; q < 4; ++q) {
        float4 hv = hp[q];
        atomicAdd(ap + q * 4 + 0, val * fmaxf(hv.x, 0.0f));
        atomicAdd(ap + q * 4 + 1, val * fmaxf(hv.y, 0.0f));
        atomicAdd(ap + q * 4 + 2, val * fmaxf(hv.z, 0.0f));
        atomicAdd(ap + q * 4 + 3, val * fmaxf(hv.w, 0.0f));
    }
}

// ---------------- contraction: logits[N,64] = af2[N,656] x W2p[656,64] via WMMA f32 ----------------
// One wave (32 lanes) per 16-node tile; 4 column tiles of 16; K stepped by 4.
// A layout (16x4 f32, ISA 7.12.2): lane<16 -> {K=0,K=1}, lane>=16 -> {K=2,K=3}, M = lane&15.
// B layout (4x16 f32, symmetric):  lane<16 -> {K=0,K=1}, lane>=16 -> {K=2,K=3}, N = lane&15.
// C/D: VGPR j holds rows {j, j+8} split across lane halves, N = lane&15.

__global__ __launch_bounds__(32) void contract_kernel(const float* __restrict__ af2,
                                                      const float* __restrict__ w2p,
                                                      float* __restrict__ logits) {
    const int tile = blockIdx.x;            // 0 .. NN/16 - 1
    const int lane = threadIdx.x & 31;
    const int m    = lane & 15;             // row within tile (A), col within tile (B/D)
    const int kh   = lane >> 4;             // K half-select

    const float* arow = af2 + ((size_t)tile * 16 + m) * KTOT;

    v8f acc0 = 0.0f, acc1 = 0.0f, acc2 = 0.0f, acc3 = 0.0f;

#pragma unroll 2
    for (int ks = 0; ks < KTOT / 4; ++ks) {
        const int k0 = ks * 4 + kh * 2;
        v2f a = *(const v2f*)(arow + k0);                    // A[m][k0], A[m][k0+1]
        const float* b0 = w2p + (size_t)k0 * NCPAD + m;      // W2p[k0][coltile*16 + m]
        const float* b1r = b0 + NCPAD;                       // W2p[k0+1][...]
        v2f b;
        b.x = b0[0];  b.y = b1r[0];
        acc0 = __builtin_amdgcn_wmma_f32_16x16x4_f32(false, a, false, b, (short)0, acc0, false, false);
        b.x = b0[16]; b.y = b1r[16];
        acc1 = __builtin_amdgcn_wmma_f32_16x16x4_f32(false, a, false, b, (short)0, acc1, false, false);
        b.x = b0[32]; b.y = b1r[32];
        acc2 = __builtin_amdgcn_wmma_f32_16x16x4_f32(false, a, false, b, (short)0, acc2, false, false);
        b.x = b0[48]; b.y = b1r[48];
        acc3 = __builtin_amdgcn_wmma_f32_16x16x4_f32(false, a, false, b, (short)0, acc3, false, false);
    }

#pragma unroll
    for (int j = 0; j < 8; ++j) {
        const size_t row = (size_t)tile * 16 + j + kh * 8;
        float* lr = logits + row * NCPAD + m;
        lr[0]  = acc0[j];
        lr[16] = acc1[j];
        lr[32] = acc2[j];
        lr[48] = acc3[j];
    }
}

// ---------------- log_softmax(logits + b2) -> out[N, 50] ----------------

__global__ void lsm_kernel(const float* __restrict__ logits, const float* __restrict__ b2,
                           float* __restrict__ out) {
    int n = blockIdx.x * blockDim.x + threadIdx.x;
    if (n >= NN) return;
    const float* lr = logits + (size_t)n * NCPAD;
    float mx = -__builtin_inff();
    for (int c = 0; c < NCLS; ++c) mx = fmaxf(mx, lr[c] + b2[c]);
    float sum = 0.0f;
    for (int c = 0; c < NCLS; ++c) sum += __expf(lr[c] + b2[c] - mx);
    float lse = __logf(sum) + mx;
    float* orow = out + (size_t)n * NCLS;
    for (int c = 0; c < NCLS; ++c) orow[c] = lr[c] + b2[c] - mx - lse;
}

// ---------------- launch ----------------

extern "C" void kernel_launch(void* const* d_in, const int* in_sizes, int n_in,
                              void* d_out, int out_size, void* d_ws, size_t ws_size,
                              hipStream_t stream) {
    const int*   src = (const int*)d_in[0];
    const int*   rel = (const int*)d_in[1];
    const int*   dst = (const int*)d_in[2];
    const float* W1  = (const float*)d_in[3];
    const float* b1  = (const float*)d_in[4];
    const float* W2  = (const float*)d_in[5];
    const float* b2  = (const float*)d_in[6];
    float* out = (float*)d_out;
    const int E = in_sizes[0];

    // workspace carve-up (floats; every section is a multiple of 4 floats -> float4 aligned)
    float* deg    = (float*)d_ws;                 // RR*NN            = 2,050,000
    float* h      = deg    + (size_t)RR * NN;     // NN*NHIDC         =   800,000
    float* w2p    = h      + (size_t)NN * NHIDC;  // KTOT*NCPAD       =    41,984
    float* logits = w2p    + (size_t)KTOT * NCPAD;// NN*NCPAD         = 3,200,000
    float* af2    = logits + (size_t)NN * NCPAD;  // NN*KTOT          = 32,800,000

    zero4_kernel<<<2048, 256, 0, stream>>>((float4*)deg, ((long long)RR * NN) / 4);
    zero4_kernel<<<8192, 256, 0, stream>>>((float4*)af2, ((long long)NN * KTOT) / 4);
    hinit_kernel<<<(NN * NHIDC + 255) / 256, 256, 0, stream>>>(h, b1, W1);
    w2pad_kernel<<<(KTOT * NCPAD + 255) / 256, 256, 0, stream>>>(w2p, W2);

    deg_kernel<<<(E + 255) / 256, 256, 0, stream>>>(src, rel, dst, deg, E);

    long long e2 = 2LL * E;
    l1_kernel<<<(unsigned)((e2 + 255) / 256), 256, 0, stream>>>(src, rel, dst, W1, deg, h, E);

    long long tot = e2 + NN;
    l2_kernel<<<(unsigned)((tot + 255) / 256), 256, 0, stream>>>(src, rel, dst, deg, h, af2, E);

    contract_kernel<<<NN / 16, 32, 0, stream>>>(af2, w2p, logits);

    lsm_kernel<<<(NN + 255) / 256, 256, 0, stream>>>(logits, b2, out);
}